// EncodeLayer_1116691497443
// MI455X (gfx1250) — compile-verified
//
#include <hip/hip_runtime.h>
#include <hip/hip_bf16.h>

// Problem constants (fixed by the reference):
//   E edges, N nodes, NH=8 heads, NT=16 key dim/head, VROW = 64*3 = 192 floats/edge value row
constexpr int NH    = 8;
constexpr int NT    = 16;
constexpr int KQROW = NH * NT;   // 128 floats per k/q row
constexpr int VROW  = 192;       // 64 channels * 3 (SO3 d)

typedef float v2f __attribute__((ext_vector_type(2)));
typedef float v4f __attribute__((ext_vector_type(4)));
typedef float v8f __attribute__((ext_vector_type(8)));

// Order-preserving f32 <-> u32 mapping so we can use global_atomic_max_u32 for
// a float segment-max (handles negatives; mxu initialized to 0 == "-inf").
__device__ __forceinline__ unsigned f2ord(float f) {
    unsigned b = __float_as_uint(f);
    return (b & 0x80000000u) ? ~b : (b | 0x80000000u);
}
__device__ __forceinline__ float ord2f(unsigned u) {
    unsigned b = (u & 0x80000000u) ? (u & 0x7fffffffu) : ~u;
    return __uint_as_float(b);
}

// ---------------------------------------------------------------------------
// Pass 1: per-edge logits via V_WMMA_F32_16X16X4_F32 + atomic segment-max.
// One wave handles 16 edges. For head h: D = A(16 edges x 16t) * B(16t x 16 edges)
// chained as 4 WMMAs of K=4; diag(D) are the 16 logits.
// Fragment layouts loaded directly from memory (ISA 7.12.2, 32-bit A 16x4 /
// B 4x16): lane L <-> edge (L&15), K-pair ((L>>4)*2) -> one float2 per fragment.
// ---------------------------------------------------------------------------
__global__ __launch_bounds__(256)
void attn_logits_wmma(const float* __restrict__ kk,
                      const float* __restrict__ qq,
                      const int*   __restrict__ dst,
                      float*       __restrict__ logits,
                      unsigned*    __restrict__ mxu,
                      int E) {
    const int lane = threadIdx.x & 31;
    const int wave = threadIdx.x >> 5;
    const int tile = blockIdx.x * (blockDim.x >> 5) + wave;
    const int e0   = tile << 4;
    if (e0 >= E) return;

    const float scale = 0.08838834764831845f;  // 1/sqrt(NH*NT) = 1/sqrt(128)

    if (e0 + 16 <= E) {                         // wave-uniform: EXEC all-1 at WMMA
        const int m16  = lane & 15;             // edge within tile for this lane
        const int hi2  = (lane >> 4) << 1;      // K-pair base: 0 or 2
        const int eIdx = e0 + m16;
        const int d    = dst[eIdx];
        const float* krow = kk + (size_t)eIdx * KQROW;
        const float* qrow = qq + (size_t)d    * KQROW;

        // Diagonal (m,m) lives in VGPR m%8 at lane m (m<8) / lane m+16 (m>=8)
        const bool active = (lane < 8) || (lane >= 24);
        const int  m      = (lane < 16) ? lane : (lane - 16);

        #pragma unroll
        for (int h = 0; h < NH; ++h) {
            v8f acc = {};
            #pragma unroll
            for (int c = 0; c < 4; ++c) {       // chain K: 4 x (16x16x4)
                const int toff = h * NT + c * 4 + hi2;
                v2f a = *(const v2f*)(krow + toff);
                v2f b = *(const v2f*)(qrow + toff);
                acc = __builtin_amdgcn_wmma_f32_16x16x4_f32(
                          false, a, false, b, (short)0, acc, false, false);
            }
            float dv = acc[0];
            #pragma unroll
            for (int j = 1; j < 8; ++j) dv = ((lane & 7) == j) ? acc[j] : dv;
            if (active) {
                const float lg = dv * scale;
                logits[(size_t)(e0 + m) * NH + h] = lg;
                atomicMax(mxu + (size_t)d * NH + h, f2ord(lg));
            }
        }
    } else {
        // scalar tail (E is a multiple of 16 in practice; kept for safety)
        const int e = e0 + lane;
        if (lane < 16 && e < E) {
            const int d = dst[e];
            for (int h = 0; h < NH; ++h) {
                float s = 0.f;
                for (int t = 0; t < NT; ++t)
                    s += kk[(size_t)e * KQROW + h * NT + t] *
                         qq[(size_t)d * KQROW + h * NT + t];
                const float lg = s * scale;
                logits[(size_t)e * NH + h] = lg;
                atomicMax(mxu + (size_t)d * NH + h, f2ord(lg));
            }
        }
    }
}

// ---------------------------------------------------------------------------
// Pass 2 (fused): ex = exp(logit - max); denom += ex; out_unnorm += ex * v.
// One thread per (edge, head): reads 24 contiguous floats of v (nontemporal,
// single-use stream) and does 24 global f32 atomic adds into L2.
// ---------------------------------------------------------------------------
__global__ __launch_bounds__(256)
void attn_scatter(const float*    __restrict__ vv,
                  const int*      __restrict__ dst,
                  const float*    __restrict__ logits,
                  const unsigned* __restrict__ mxu,
                  float*          __restrict__ denom,
                  float*          __restrict__ out,
                  int E) {
    const int tid = blockIdx.x * blockDim.x + threadIdx.x;
    if (tid >= E * NH) return;
    const int e = tid >> 3;
    const int h = tid & 7;
    const int d = dst[e];

    const float mx = ord2f(mxu[(size_t)d * NH + h]);
    const float ex = expf(logits[tid] - mx);
    atomicAdd(denom + (size_t)d * NH + h, ex);

    const v4f* vp = (const v4f*)(vv + (size_t)e * VROW + h * 24);
    float*     op = out + (size_t)d * VROW + h * 24;
    #pragma unroll
    for (int j = 0; j < 6; ++j) {
        v4f val = __builtin_nontemporal_load(vp + j);
        atomicAdd(op + j * 4 + 0, ex * val.x);
        atomicAdd(op + j * 4 + 1, ex * val.y);
        atomicAdd(op + j * 4 + 2, ex * val.z);
        atomicAdd(op + j * 4 + 3, ex * val.w);
    }
}

// ---------------------------------------------------------------------------
// Pass 3: out = out_unnorm / denom (0 for empty segments, matching reference).
// ---------------------------------------------------------------------------
__global__ __launch_bounds__(256)
void attn_norm(float* __restrict__ out, const float* __restrict__ denom, int N) {
    const int i = blockIdx.x * blockDim.x + threadIdx.x;
    if (i >= N * VROW) return;
    const int n  = i / VROW;
    const int h  = (i % VROW) / 24;
    const float dn  = denom[(size_t)n * NH + h];
    const float val = out[i];
    out[i] = (dn > 0.f) ? (val / dn) : 0.f;
}

extern "C" void kernel_launch(void* const* d_in, const int* in_sizes, int n_in,
                              void* d_out, int out_size, void* d_ws, size_t ws_size,
                              hipStream_t stream) {
    const float* v   = (const float*)d_in[0];   // [E, 64, 3]
    const float* k   = (const float*)d_in[1];   // [E, 8, 16]
    const float* q   = (const float*)d_in[2];   // [N, 8, 16]
    const int*   dst = (const int*)d_in[3];     // [E]
    const int E = in_sizes[3];
    const int N = in_sizes[2] / KQROW;
    float* out = (float*)d_out;

    // Workspace layout: logits[E*8] f32 | mxu[N*8] u32 | denom[N*8] f32
    float*    logits = (float*)d_ws;
    unsigned* mxu    = (unsigned*)((char*)d_ws + (size_t)E * NH * sizeof(float));
    float*    denom  = (float*)((char*)mxu + (size_t)N * NH * sizeof(unsigned));

    // Init accumulators each call (capture-safe async memsets).
    hipMemsetAsync(d_out, 0, (size_t)out_size * sizeof(float), stream);
    hipMemsetAsync(mxu, 0, (size_t)N * NH * (sizeof(unsigned) + sizeof(float)), stream);

    const int tiles = (E + 15) / 16;                  // 16 edges per wave
    attn_logits_wmma<<<dim3((tiles + 7) / 8), dim3(256), 0, stream>>>(
        k, q, dst, logits, mxu, E);

    const long long w2 = (long long)E * NH;
    attn_scatter<<<dim3((unsigned)((w2 + 255) / 256)), dim3(256), 0, stream>>>(
        v, dst, logits, mxu, denom, out, E);

    const long long w3 = (long long)N * VROW;
    attn_norm<<<dim3((unsigned)((w3 + 255) / 256)), dim3(256), 0, stream>>>(
        out, denom, N);
}